// MultiResolutionHashGrid_71863392796822
// MI455X (gfx1250) — compile-verified
//
#include <hip/hip_runtime.h>
#include <math.h>

// Multi-resolution hash grid encode (no interpolation), MI455X / gfx1250.
//
// Per (point p, level l):
//   xn = (x+1)/2 ; s = xn*(res_l - 1) ; g = trunc(s) (floor, s>=0)
//   h = g.x ^ g.y*P1 ^ g.z*P2 ; idx = h & (2^19 - 1)
//   out[p*32 + l*2 .. +1] = table_l[idx*2 .. +1]
//
// Strategy: memory-bound gather. Tables (64 MB total) fit in the 192 MB L2;
// the 256 MB output is streamed with NON-TEMPORAL stores so it does not evict
// the tables. One wave32 lane per (p,l) pair: per-wave stores cover 256
// contiguous bytes, gathers are one independent b64 per lane.

#define N_LEVELS   16
#define LOG2_T     19
#define TABLE_MASK ((1u << LOG2_T) - 1u)
#define P1_HASH    2654435761u
#define P2_HASH    805459861u

typedef __attribute__((ext_vector_type(2))) float v2f;

struct ResM1 { float v[N_LEVELS]; };   // (float)resolution[l] - 1.0f

__global__ __launch_bounds__(256) void hashgrid_encode_kernel(
    const float* __restrict__ x,        // [N, 3]
    const float* __restrict__ tables,   // [16, 2^19, 2]
    float* __restrict__ out,            // [N, 32]
    ResM1 resm1,
    unsigned total)                     // N * 16
{
  unsigned t = blockIdx.x * blockDim.x + threadIdx.x;
  if (t >= total) return;

  unsigned p = t >> 4;        // point index
  unsigned l = t & 15u;       // level index

  // 12-byte point read; broadcast across the 16 lanes sharing p (L0 hit).
  const float* xp = x + (size_t)p * 3u;
  float x0 = xp[0];
  float x1 = xp[1];
  float x2 = xp[2];

  // Match reference rounding order: x_norm rounded first, then scaled.
  float s = resm1.v[l];
  float n0 = (x0 + 1.0f) * 0.5f;
  float n1 = (x1 + 1.0f) * 0.5f;
  float n2 = (x2 + 1.0f) * 0.5f;
  unsigned g0 = (unsigned)(n0 * s);   // trunc == floor (non-negative)
  unsigned g1 = (unsigned)(n1 * s);
  unsigned g2 = (unsigned)(n2 * s);

  unsigned h   = g0 ^ (g1 * P1_HASH) ^ (g2 * P2_HASH);
  unsigned idx = h & TABLE_MASK;

  // Gather 2 features (8 bytes) from level l's table; default (RT) temporal
  // hint keeps the 64 MB table set resident in L2.
  const v2f* tbl = (const v2f*)(tables + (((size_t)l << LOG2_T) << 1));
  v2f feat = tbl[idx];

  // Streamed output: non-temporal store (th:TH_STORE_NT) so the 256 MB
  // output stream does not evict the L2-resident tables. Lanes 0..15 of the
  // half-wave cover 128 contiguous bytes -> fully coalesced b64 stores.
  v2f* dst = (v2f*)(out + ((size_t)p << 5) + ((size_t)l << 1));
  __builtin_nontemporal_store(feat, dst);
}

extern "C" void kernel_launch(void* const* d_in, const int* in_sizes, int n_in,
                              void* d_out, int out_size, void* d_ws, size_t ws_size,
                              hipStream_t stream) {
  (void)n_in; (void)out_size; (void)d_ws; (void)ws_size;

  const float* x      = (const float*)d_in[0];   // [N,3] f32
  const float* tables = (const float*)d_in[1];   // [16, 2^19, 2] f32
  float*       out    = (float*)d_out;           // [N,32] f32

  unsigned n     = (unsigned)(in_sizes[0] / 3);
  unsigned total = n * (unsigned)N_LEVELS;

  // Reproduce the harness's resolution table bit-exactly: CPython's
  // math.exp/math.log and float**int all call the same glibc
  // exp/log/pow doubles we call here, so int truncation matches even at the
  // 16*b^15 ~= 2048 boundary.
  ResM1 r;
  double b = exp((log(2048.0) - log(16.0)) / 15.0);
  for (int l = 0; l < N_LEVELS; ++l) {
    int res = (int)(16.0 * pow(b, (double)l));
    r.v[l] = (float)res - 1.0f;
  }

  dim3 block(256);
  dim3 grid((total + 255u) / 256u);
  hipLaunchKernelGGL(hashgrid_encode_kernel, grid, block, 0, stream,
                     x, tables, out, r, total);
}